// EnhancedHDE_DGCNN_35235911696999
// MI455X (gfx1250) — compile-verified
//
#include <hip/hip_runtime.h>
#include <math.h>

#define NN 50000
#define NE 800000
#define NG 100
#define NPG 500
#define HID 64
#define KTOP 15
#define CATD 193
#define POOLD 2895
#define LDSP 66  // padded LDS stride for transposed W (conflict-free ds_load_b64)

typedef __attribute__((ext_vector_type(2))) float v2f;
typedef __attribute__((ext_vector_type(8))) float v8f;

// ---------------- degree / norm ----------------
__global__ void k_deg_init(float* deg, int n) {
  int i = blockIdx.x * blockDim.x + threadIdx.x;
  if (i < n) deg[i] = 1.0f;  // self loop
}
__global__ void k_deg_count(const int* __restrict__ dst, float* deg, int e) {
  int i = blockIdx.x * blockDim.x + threadIdx.x;
  if (i < e) atomicAdd(&deg[dst[i]], 1.0f);
}
__global__ void k_dinv(float* d, int n) {
  int i = blockIdx.x * blockDim.x + threadIdx.x;
  if (i < n) d[i] = 1.0f / sqrtf(d[i]);  // deg >= 1 always
}

// ---------------- specialized GCN GEMM: C[M,64] = A[M,64] @ W[64,64] ----------------
// One wave computes a 16x64 row stripe (4 WMMA tiles). K fully unrolled, no guards.
// W is staged transposed in padded LDS so B fragments are single ds_load_b64's.
__global__ void __launch_bounds__(256) k_gcn_gemm(const float* __restrict__ A,
                                                  const float* __restrict__ W,
                                                  float* __restrict__ C,
                                                  int nRowTiles) {
  __shared__ float WT[64 * LDSP];  // WT[col*LDSP + k]
  const int tid = threadIdx.x;
  // cooperative load + transpose of W (64x64): coalesced global reads
  for (int i = tid; i < 4096; i += 256) {
    int r = i >> 6, cl = i & 63;
    WT[cl * LDSP + r] = W[i];
  }
  __syncthreads();

  const int lane = tid & 31;
  const int wave = tid >> 5;
  const int tile = blockIdx.x * 8 + wave;
  if (tile >= nRowTiles) return;  // wave-uniform
  const int row0 = tile << 4;
  const int half = lane >> 4;  // 0 -> K={0,1}; 1 -> K={2,3} of each x4 step
  const int l16 = lane & 15;
  const float* arowp = A + (size_t)(row0 + l16) * 64;
  const float* wb0 = &WT[(l16 + 0) * LDSP];
  const float* wb1 = &WT[(l16 + 16) * LDSP];
  const float* wb2 = &WT[(l16 + 32) * LDSP];
  const float* wb3 = &WT[(l16 + 48) * LDSP];

  v8f c0 = {}, c1 = {}, c2 = {}, c3 = {};
#pragma unroll
  for (int kk = 0; kk < 16; ++kk) {
    const int kb = (kk << 2) + (half << 1);  // even -> 8B aligned
    v2f a  = *reinterpret_cast<const v2f*>(arowp + kb);
    v2f b0 = *reinterpret_cast<const v2f*>(wb0 + kb);
    v2f b1 = *reinterpret_cast<const v2f*>(wb1 + kb);
    v2f b2 = *reinterpret_cast<const v2f*>(wb2 + kb);
    v2f b3 = *reinterpret_cast<const v2f*>(wb3 + kb);
    c0 = __builtin_amdgcn_wmma_f32_16x16x4_f32(false, a, false, b0, (short)0, c0, false, false);
    c1 = __builtin_amdgcn_wmma_f32_16x16x4_f32(false, a, false, b1, (short)0, c1, false, false);
    c2 = __builtin_amdgcn_wmma_f32_16x16x4_f32(false, a, false, b2, (short)0, c2, false, false);
    c3 = __builtin_amdgcn_wmma_f32_16x16x4_f32(false, a, false, b3, (short)0, c3, false, false);
  }
#pragma unroll
  for (int r = 0; r < 8; ++r) {
    float* cp = C + (size_t)(row0 + (half << 3) + r) * 64;
    cp[l16 +  0] = c0[r];
    cp[l16 + 16] = c1[r];
    cp[l16 + 32] = c2[r];
    cp[l16 + 48] = c3[r];
  }
}

// ---------------- generic branchless WMMA GEMM (classifier) ----------------
// C[M,N] = A[M,K] @ B[K,N] (+bias)(+BN)(+ReLU). N multiple of 16.
// Out-of-range handled via clamped-index loads * {0,1} masks (no exec branching).
__global__ void k_gemm_wmma(const float* __restrict__ A, int lda,
                            const float* __restrict__ B, int ldb,
                            const float* __restrict__ bias,
                            const float* __restrict__ gamma,
                            const float* __restrict__ beta,
                            float bnscale,
                            float* __restrict__ C, int ldc,
                            int M, int N, int K, int act) {
  const int lane = threadIdx.x & 31;
  const int wave = threadIdx.x >> 5;
  const int nTilesN = N >> 4;
  const int nTilesM = (M + 15) >> 4;
  const int tile = blockIdx.x * (blockDim.x >> 5) + wave;
  if (tile >= nTilesM * nTilesN) return;  // wave-uniform
  const int rowTile = tile / nTilesN;
  const int colTile = tile - rowTile * nTilesN;
  const int row0 = rowTile << 4;
  const int col0 = colTile << 4;
  const int half = lane >> 4;
  const int l16 = lane & 15;
  const int col = col0 + l16;
  const int arow = row0 + l16;
  const int ra = arow < M ? arow : M - 1;      // clamped (always in-bounds)
  const float am = (arow < M) ? 1.0f : 0.0f;   // row validity mask
  const float* ap = A + (size_t)ra * lda;

  v8f c = {};
  const int K4 = K >> 2;  // full steps, no k-guard
  for (int kk = 0; kk < K4; ++kk) {
    const int kb = (kk << 2) + (half << 1);
    v2f a, b;
    a.x = ap[kb] * am;
    a.y = ap[kb + 1] * am;
    b.x = B[(size_t)kb * ldb + col];
    b.y = B[(size_t)(kb + 1) * ldb + col];
    c = __builtin_amdgcn_wmma_f32_16x16x4_f32(false, a, false, b, (short)0, c, false, false);
  }
  if (K & 3) {  // masked tail step
    const int kb = (K4 << 2) + (half << 1);
    const int k0 = kb < K ? kb : K - 1;
    const int k1 = kb + 1 < K ? kb + 1 : K - 1;
    const float m0 = (kb < K) ? 1.0f : 0.0f;
    const float m1 = (kb + 1 < K) ? 1.0f : 0.0f;
    v2f a, b;
    a.x = ap[k0] * (am * m0);
    a.y = ap[k1] * (am * m1);
    b.x = B[(size_t)k0 * ldb + col] * m0;
    b.y = B[(size_t)k1 * ldb + col] * m1;
    c = __builtin_amdgcn_wmma_f32_16x16x4_f32(false, a, false, b, (short)0, c, false, false);
  }
#pragma unroll
  for (int r = 0; r < 8; ++r) {
    int row = row0 + (half << 3) + r;
    if (row < M) {
      float y = c[r];
      if (bias)  y += bias[col];
      if (gamma) y = y * bnscale * gamma[col] + beta[col];
      if (act == 1) y = fmaxf(y, 0.0f);
      C[(size_t)row * ldc + col] = y;
    }
  }
}

// ---------------- GCN scatter pieces ----------------
__global__ void k_selfinit(const float* __restrict__ lin,
                           const float* __restrict__ dinv,
                           float* acc, int total, int shift) {
  int i = blockIdx.x * blockDim.x + threadIdx.x;
  if (i < total) {
    float w = dinv[i >> shift];
    acc[i] = lin[i] * w * w;
  }
}

__global__ void k_scatter64(const int* __restrict__ src, const int* __restrict__ dst,
                            const float* __restrict__ dinv,
                            const float* __restrict__ lin,
                            float* acc, int e) {
  int t = blockIdx.x * blockDim.x + threadIdx.x;
  int edge = t >> 4;
  if (edge >= e) return;
  int c = (t & 15) << 2;
  int s = src[edge], d = dst[edge];
  float w = dinv[s] * dinv[d];
  const float4 hv = *reinterpret_cast<const float4*>(lin + s * 64 + c);
  float* out = acc + d * 64 + c;
  atomicAdd(out + 0, hv.x * w);
  atomicAdd(out + 1, hv.y * w);
  atomicAdd(out + 2, hv.z * w);
  atomicAdd(out + 3, hv.w * w);
}

__global__ void k_scatter1(const int* __restrict__ src, const int* __restrict__ dst,
                           const float* __restrict__ dinv,
                           const float* __restrict__ lin1,
                           float* acc1, int e) {
  int i = blockIdx.x * blockDim.x + threadIdx.x;
  if (i < e) {
    int s = src[i], d = dst[i];
    atomicAdd(&acc1[d], lin1[s] * dinv[s] * dinv[d]);
  }
}

__global__ void k_finalize(const float* __restrict__ acc,
                           const float* __restrict__ bias,
                           float* h, int total, int fmask) {
  int i = blockIdx.x * blockDim.x + threadIdx.x;
  if (i < total) h[i] = tanhf(acc[i] + bias[i & fmask]);
}

__global__ void k_dense64(const float* __restrict__ A, const float* __restrict__ w,
                          float* out, int n) {
  int v = blockIdx.x * blockDim.x + threadIdx.x;
  if (v < n) {
    const float4* a = reinterpret_cast<const float4*>(A + (size_t)v * 64);
    const float4* wv = reinterpret_cast<const float4*>(w);
    float s = 0.0f;
#pragma unroll
    for (int j = 0; j < 16; ++j) {
      float4 av = a[j], wj = wv[j];
      s += av.x * wj.x + av.y * wj.y + av.z * wj.z + av.w * wj.w;
    }
    out[v] = s;
  }
}

// ---------------- top-K sort pool (one block / graph) ----------------
__global__ void k_topk_pool(const float* __restrict__ h0, const float* __restrict__ h1,
                            const float* __restrict__ h2, const float* __restrict__ h3,
                            float* __restrict__ pooled) {
  __shared__ float score[NPG];
  __shared__ float sval[512];
  __shared__ int   sidx[512];
  __shared__ int   topidx[KTOP];
  const int g = blockIdx.x;
  const int t = threadIdx.x;  // 512 threads
  for (int j = t; j < NPG; j += 512) score[j] = h3[g * NPG + j];
  __syncthreads();
  for (int k = 0; k < KTOP; ++k) {
    float v = (t < NPG) ? score[t] : -3.0f;  // real scores are tanh in [-1,1]
    sval[t] = v;
    sidx[t] = t;
    __syncthreads();
    for (int off = 256; off > 0; off >>= 1) {
      if (t < off) {
        float v2 = sval[t + off];
        int i2 = sidx[t + off];
        if (v2 > sval[t] || (v2 == sval[t] && i2 < sidx[t])) {
          sval[t] = v2;
          sidx[t] = i2;
        }
      }
      __syncthreads();
    }
    if (t == 0) {
      topidx[k] = sidx[0];
      score[sidx[0]] = -2.0f;
    }
    __syncthreads();
  }
  for (int i = t; i < POOLD; i += 512) {
    int k = i / CATD;
    int c = i - k * CATD;
    int node = g * NPG + topidx[k];
    float val;
    if (c < 64)       val = h0[node * 64 + c];
    else if (c < 128) val = h1[node * 64 + (c - 64)];
    else if (c < 192) val = h2[node * 64 + (c - 128)];
    else              val = h3[node];
    pooled[g * POOLD + i] = val;
  }
}

__global__ void k_final(const float* __restrict__ Z, const float* __restrict__ w,
                        const float* __restrict__ b, float* out, int m) {
  int r = blockIdx.x * blockDim.x + threadIdx.x;
  if (r < m) {
    float s = b[0];
#pragma unroll
    for (int j = 0; j < 64; ++j) s += Z[r * 64 + j] * w[j];
    out[r] = s;
  }
}

extern "C" void kernel_launch(void* const* d_in, const int* in_sizes, int n_in,
                              void* d_out, int out_size, void* d_ws, size_t ws_size,
                              hipStream_t stream) {
  const float* x   = (const float*)d_in[0];
  const int*   ei  = (const int*)d_in[1];
  const int*   src = ei;
  const int*   dst = ei + NE;
  const float* W[4] = {(const float*)d_in[3], (const float*)d_in[5],
                       (const float*)d_in[7], (const float*)d_in[9]};
  const float* b[4] = {(const float*)d_in[4], (const float*)d_in[6],
                       (const float*)d_in[8], (const float*)d_in[10]};
  const float* Wc0 = (const float*)d_in[11]; const float* bc0 = (const float*)d_in[12];
  const float* Wc1 = (const float*)d_in[13]; const float* bc1 = (const float*)d_in[14];
  const float* Wc2 = (const float*)d_in[15]; const float* bc2 = (const float*)d_in[16];
  const float* Wc3 = (const float*)d_in[17]; const float* bc3 = (const float*)d_in[18];
  const float* gamma = (const float*)d_in[19];
  const float* beta  = (const float*)d_in[20];
  float* out = (float*)d_out;

  // ---- workspace carve (floats) ----
  float* ws    = (float*)d_ws;
  float* dinv  = ws;                   ws += NN;
  float* lin   = ws;                   ws += (size_t)NN * 64;
  float* acc   = ws;                   ws += (size_t)NN * 64;
  float* h0    = ws;                   ws += (size_t)NN * 64;
  float* h1    = ws;                   ws += (size_t)NN * 64;
  float* h2    = ws;                   ws += (size_t)NN * 64;
  float* h3    = ws;                   ws += NN;
  float* lin1  = ws;                   ws += NN;
  float* acc1  = ws;                   ws += NN;
  float* pooled= ws;                   ws += (size_t)NG * POOLD;
  float* z0    = ws;                   ws += (size_t)NG * 256;
  float* z1    = ws;                   ws += (size_t)NG * 128;
  float* z2    = ws;                   ws += (size_t)NG * 64;

  const float bnscale = 1.0f / sqrtf(1.0f + 1e-5f);

  // ---- degrees / normalization ----
  k_deg_init<<<(NN + 255) / 256, 256, 0, stream>>>(dinv, NN);
  k_deg_count<<<(NE + 255) / 256, 256, 0, stream>>>(dst, dinv, NE);
  k_dinv<<<(NN + 255) / 256, 256, 0, stream>>>(dinv, NN);

  // ---- GCN layers 0..2 (64 -> 64) ----
  const float* prev = x;
  float* houts[3] = {h0, h1, h2};
  const int nRowTiles = NN / 16;                       // 3125
  const int gemmBlocks = (nRowTiles + 7) / 8;          // 8 waves / block
  for (int l = 0; l < 3; ++l) {
    k_gcn_gemm<<<gemmBlocks, 256, 0, stream>>>(prev, W[l], lin, nRowTiles);
    k_selfinit<<<(NN * 64 + 255) / 256, 256, 0, stream>>>(lin, dinv, acc, NN * 64, 6);
    k_scatter64<<<(NE * 16 + 255) / 256, 256, 0, stream>>>(src, dst, dinv, lin, acc, NE);
    k_finalize<<<(NN * 64 + 255) / 256, 256, 0, stream>>>(acc, b[l], houts[l], NN * 64, 63);
    prev = houts[l];
  }

  // ---- GCN layer 3 (64 -> 1) ----
  k_dense64<<<(NN + 255) / 256, 256, 0, stream>>>(h2, W[3], lin1, NN);
  k_selfinit<<<(NN + 255) / 256, 256, 0, stream>>>(lin1, dinv, acc1, NN, 0);
  k_scatter1<<<(NE + 255) / 256, 256, 0, stream>>>(src, dst, dinv, lin1, acc1, NE);
  k_finalize<<<(NN + 255) / 256, 256, 0, stream>>>(acc1, b[3], h3, NN, 0);

  // ---- global sort pool ----
  k_topk_pool<<<NG, 512, 0, stream>>>(h0, h1, h2, h3, pooled);

  // ---- classifier ----
  {  // L0: 100 x 2895 @ 2895 x 256, bias + BN + ReLU
    int tiles = ((NG + 15) / 16) * (256 / 16);  // 112
    k_gemm_wmma<<<(tiles + 7) / 8, 256, 0, stream>>>(pooled, POOLD, Wc0, 256,
        bc0, gamma, beta, bnscale, z0, 256, NG, 256, POOLD, 1);
  }
  {  // L1: 100 x 256 @ 256 x 128, bias + ReLU
    int tiles = ((NG + 15) / 16) * (128 / 16);  // 56
    k_gemm_wmma<<<(tiles + 7) / 8, 256, 0, stream>>>(z0, 256, Wc1, 128,
        bc1, nullptr, nullptr, 1.0f, z1, 128, NG, 128, 256, 1);
  }
  {  // L2: 100 x 128 @ 128 x 64, bias + ReLU
    int tiles = ((NG + 15) / 16) * (64 / 16);   // 28
    k_gemm_wmma<<<(tiles + 7) / 8, 256, 0, stream>>>(z1, 128, Wc2, 64,
        bc2, nullptr, nullptr, 1.0f, z2, 64, NG, 64, 128, 1);
  }
  // L3: 100 x 64 @ 64 x 1 + bias -> logits
  k_final<<<1, 128, 0, stream>>>(z2, Wc3, bc3, out, NG);
}